// GnnCapsuleLayer_89747636617734
// MI455X (gfx1250) — compile-verified
//
#include <hip/hip_runtime.h>

typedef __attribute__((ext_vector_type(2))) float v2f;
typedef __attribute__((ext_vector_type(4))) float v4f;
typedef __attribute__((ext_vector_type(8))) float v8f;

#define NN 36
#define CC 8
#define GPB 32                               // graphs per block
#define GPW 4                                // graphs per wave (8 waves/block)
#define FLOATS_PER_BLOCK (GPB * NN * CC)     // 9216 floats = 36 KB
#define VECS_PER_BLOCK (FLOATS_PER_BLOCK / 4)
#define VECS_PER_THREAD (VECS_PER_BLOCK / 256)   // 9

__global__ __launch_bounds__(256)
void sage_fused_wmma(const float* __restrict__ x, const float* __restrict__ Wl,
                     const float* __restrict__ bl, const float* __restrict__ Wr,
                     float* __restrict__ out, int nGraphs)
{
    __shared__ float xs[FLOATS_PER_BLOCK];   // 32 graphs of x; overwritten in place with out
    __shared__ float Ss[GPB * CC];           // per-graph channel sums

    const int tid = threadIdx.x;
    const long long fbase  = (long long)blockIdx.x * FLOATS_PER_BLOCK;
    const long long flimit = (long long)nGraphs * (NN * CC);
    const bool full = (fbase + FLOATS_PER_BLOCK) <= flimit;   // uniform per block

    // ---- global -> LDS: async DMA fast path (no VGPR round-trip, 9 copies in flight) ----
    if (full) {
#pragma unroll
        for (int it = 0; it < VECS_PER_THREAD; ++it) {
            const int j = tid + it * 256;
            const unsigned lds_off = (unsigned)(uintptr_t)(xs + j * 4);
            const float* gp = x + fbase + (long long)j * 4;
            asm volatile("global_load_async_to_lds_b128 %0, %1, off"
                         :: "v"(lds_off), "v"(gp) : "memory");
        }
        asm volatile("s_wait_asynccnt 0x0" ::: "memory");
    } else {
        for (int j = tid; j < VECS_PER_BLOCK; j += 256) {
            long long fidx = fbase + (long long)j * 4;
            v4f v = {};
            if (fidx + 3 < flimit) v = *(const v4f*)(x + fidx);
            *(v4f*)(xs + j * 4) = v;
        }
    }
    __syncthreads();

    // ---- per-graph channel sums: thread t -> graph t>>3, channel t&7 ----
    {
        const int g = tid >> 3, c = tid & 7;
        float s = 0.f;
        const float* p = xs + g * (NN * CC) + c;
#pragma unroll
        for (int i = 0; i < NN; ++i) s += p[i * CC];
        Ss[tid] = s;
    }

    // ---- build uniform B fragments + bias (per lane) ----
    // B (16x16): rows 0..7 = W_r^T - W_l^T/35 ; rows 8..15 = W_l^T/35 ; cols 8..15 = 0
    // f32 B layout (mirror of documented A layout): VGPR0 = K even, VGPR1 = K odd,
    // lanes 16..31 hold K+2 for the same N = lane&15.
    const int lane = tid & 31;
    const int half = lane >> 4;
    const int n    = lane & 15;
    const float inv35 = 1.0f / 35.0f;
    v2f bfrag[4];
#pragma unroll
    for (int k = 0; k < 4; ++k) {
        const int kk0 = 4 * k + (half ? 2 : 0);
        float b0 = 0.f, b1 = 0.f;
        if (n < 8) {
            if (k < 2) {   // kk in 0..7
                b0 = Wr[n * CC + kk0]     - Wl[n * CC + kk0]     * inv35;
                b1 = Wr[n * CC + kk0 + 1] - Wl[n * CC + kk0 + 1] * inv35;
            } else {       // kk in 8..15
                const int c0 = kk0 - 8;
                b0 = Wl[n * CC + c0]     * inv35;
                b1 = Wl[n * CC + c0 + 1] * inv35;
            }
        }
        v2f bf; bf.x = b0; bf.y = b1;
        bfrag[k] = bf;
    }
    const float bias = bl[n & 7];

    __syncthreads();

    // ---- WMMA: each wave handles 4 whole graphs = 144 rows = 9 tiles of 16 rows ----
    const int w = tid >> 5;
    const int rowBase = w * (GPW * NN);
#pragma unroll 1
    for (int t = 0; t < (GPW * NN) / 16; ++t) {
        v8f c;
#pragma unroll
        for (int v = 0; v < 8; ++v) c[v] = bias;   // bias folded into accumulator (col = lane&15)

        const int row = rowBase + t * 16 + n;      // A layout: M = lane&15 for both halves
        const int g   = row / NN;                  // local graph owning this row
        const float* xr = xs + row * CC;
        const float* sr = Ss + g * CC;
        const int off = half ? 2 : 0;
        const v2f a0 = *(const v2f*)(xr + 0 + off);  // K 0..3
        const v2f a1 = *(const v2f*)(xr + 4 + off);  // K 4..7
        const v2f a2 = *(const v2f*)(sr + 0 + off);  // K 8..11  (S part)
        const v2f a3 = *(const v2f*)(sr + 4 + off);  // K 12..15 (S part)

        c = __builtin_amdgcn_wmma_f32_16x16x4_f32(false, a0, false, bfrag[0], (short)0, c, false, false);
        c = __builtin_amdgcn_wmma_f32_16x16x4_f32(false, a1, false, bfrag[1], (short)0, c, false, false);
        c = __builtin_amdgcn_wmma_f32_16x16x4_f32(false, a2, false, bfrag[2], (short)0, c, false, false);
        c = __builtin_amdgcn_wmma_f32_16x16x4_f32(false, a3, false, bfrag[3], (short)0, c, false, false);

        // D layout: VGPR v, lanes 0-15 -> row v, lanes 16-31 -> row v+8; col = lane&15.
        // Only cols 0..7 are valid; write back in place over this tile's x rows.
        if (n < 8) {
            float* orow = xs + (rowBase + t * 16 + half * 8) * CC + n;
#pragma unroll
            for (int v = 0; v < 8; ++v) orow[v * CC] = c[v];
        }
    }
    __syncthreads();

    // ---- LDS -> global: async DMA fast path; S_ENDPGM's implicit wait-idle drains ASYNCcnt ----
    if (full) {
#pragma unroll
        for (int it = 0; it < VECS_PER_THREAD; ++it) {
            const int j = tid + it * 256;
            const unsigned lds_off = (unsigned)(uintptr_t)(xs + j * 4);
            float* gp = out + fbase + (long long)j * 4;
            asm volatile("global_store_async_from_lds_b128 %0, %1, off"
                         :: "v"(gp), "v"(lds_off) : "memory");
        }
    } else {
        for (int j = tid; j < VECS_PER_BLOCK; j += 256) {
            long long fidx = fbase + (long long)j * 4;
            if (fidx + 3 < flimit) *(v4f*)(out + fidx) = *(const v4f*)(xs + j * 4);
        }
    }
}

extern "C" void kernel_launch(void* const* d_in, const int* in_sizes, int n_in,
                              void* d_out, int out_size, void* d_ws, size_t ws_size,
                              hipStream_t stream) {
    (void)n_in; (void)out_size; (void)d_ws; (void)ws_size;
    const float* x  = (const float*)d_in[0];
    const float* Wl = (const float*)d_in[1];
    const float* bl = (const float*)d_in[2];
    const float* Wr = (const float*)d_in[3];
    float* out = (float*)d_out;
    const int nGraphs = in_sizes[0] / (NN * CC);
    const int blocks = (nGraphs + GPB - 1) / GPB;
    sage_fused_wmma<<<blocks, 256, 0, stream>>>(x, Wl, bl, Wr, out, nGraphs);
}